// CNTF_83683142795463
// MI455X (gfx1250) — compile-verified
//
#include <hip/hip_runtime.h>

#define NPAT 8
#define NNZ 400000
#define TDIM 512
#define LDIM 4096
#define MDIM 4096
#define RDIM 64
#define ITEMS 16
#define BLK 256

typedef __attribute__((ext_vector_type(2))) float v2f;
typedef __attribute__((ext_vector_type(4))) float v4f;
typedef __attribute__((ext_vector_type(8))) float v8f;
typedef __attribute__((ext_vector_type(4))) int v4i;

// ---------------------------------------------------------------------------
// Kernel 1: column sums via V_WMMA_F32_16X16X4_F32 with all-ones A matrix.
// D[m,n] = sum_k A[m,k]*B[k,n] + C; A==1 -> every D row = column sums of B.
// Jobs: 0 = Ul (4096xR), 1 = Um (4096xR), 2..9 = Ws[p] (512xR).
// Block = 128 threads (4 waves); wave w owns columns [16w, 16w+16).
// ---------------------------------------------------------------------------
__global__ __launch_bounds__(128) void colsum_wmma(
    const float* __restrict__ Ul, const float* __restrict__ Um,
    const float* __restrict__ Ws, float* __restrict__ sums /* [10][RDIM] */) {
  const int job = blockIdx.x;
  const float* src;
  int rows;
  if (job == 0) { src = Ul; rows = LDIM; }
  else if (job == 1) { src = Um; rows = MDIM; }
  else { src = Ws + (size_t)(job - 2) * TDIM * RDIM; rows = TDIM; }

  const int lane = threadIdx.x & 31;
  const int wv = threadIdx.x >> 5;
  const int n = wv * 16 + (lane & 15);      // output column this lane holds in B/C/D
  const int kb = (lane >> 4) << 1;          // K rows 0,1 (lanes 0-15) or 2,3 (16-31)

  v2f a; a.x = 1.0f; a.y = 1.0f;            // A = ones(16x4)
  v8f c = {};
  for (int r0 = 0; r0 < rows; r0 += 4) {
    v2f b;
    b.x = src[(size_t)(r0 + kb + 0) * RDIM + n];   // B VGPR0: K = kb
    b.y = src[(size_t)(r0 + kb + 1) * RDIM + n];   // B VGPR1: K = kb+1
    c = __builtin_amdgcn_wmma_f32_16x16x4_f32(
        /*neg_a=*/false, a, /*neg_b=*/false, b,
        /*c_mod=*/(short)0, c, /*reuse_a=*/false, /*reuse_b=*/false);
  }
  // D row M=0 lives in c[0] on lanes 0..15 (N = lane).
  if (lane < 16) sums[job * RDIM + n] = c[0];
}

// ---------------------------------------------------------------------------
// Kernel 2: out = sum_p sum_M_p / T   (also initializes d_out, overwriting
// the 0xAA poison; gather kernel below then atomically subtracts).
// One wave of 32 lanes; lane handles r and r+32.
// ---------------------------------------------------------------------------
__global__ __launch_bounds__(32) void summ_kernel(const float* __restrict__ sums,
                                                  float* __restrict__ out) {
  const int lane = threadIdx.x;
  const float t0 = sums[0 * RDIM + lane] * sums[1 * RDIM + lane];
  const float t1 = sums[0 * RDIM + lane + 32] * sums[1 * RDIM + lane + 32];
  float tot = 0.0f;
#pragma unroll
  for (int p = 0; p < NPAT; ++p) {
    tot += sums[(2 + p) * RDIM + lane] * t0 +
           sums[(2 + p) * RDIM + lane + 32] * t1;
  }
#pragma unroll
  for (int off = 16; off > 0; off >>= 1) tot += __shfl_down(tot, off, 32);
  if (lane == 0) out[0] = tot / (float)TDIM;
}

// ---------------------------------------------------------------------------
// Kernel 3: main gather. grid = (ceil(NNZ / (BLK*ITEMS)), NPAT).
// Stage Wp (128 KB) in LDS (async global->LDS when available), then per-nnz
// triple gather + R-reduction + v_log_f32, block reduce, one atomicAdd.
// ---------------------------------------------------------------------------
__global__ __launch_bounds__(BLK) void gather_ll(
    const int* __restrict__ s0, const int* __restrict__ s1,
    const int* __restrict__ s2, const float* __restrict__ vals,
    const float* __restrict__ Ws, const float* __restrict__ Ul,
    const float* __restrict__ Um, float* __restrict__ out) {
  __shared__ float sWp[TDIM * RDIM];  // 128 KB (<= 320 KB/workgroup on CDNA5)
  __shared__ float wsum[BLK / 32];

  const int p = blockIdx.y;
  const float* Wp = Ws + (size_t)p * TDIM * RDIM;
  const int nvec = TDIM * RDIM / 4;  // 8192 x b128

#if defined(__gfx1250__) && __has_builtin(__builtin_amdgcn_global_load_async_to_lds_b128) && __has_builtin(__builtin_amdgcn_s_wait_asynccnt)
  {
    auto gsrc = (__attribute__((address_space(1))) v4i*)Wp;
    auto ldst = (__attribute__((address_space(3))) v4i*)sWp;
    for (int i = threadIdx.x; i < nvec; i += BLK) {
      __builtin_amdgcn_global_load_async_to_lds_b128(gsrc + i, ldst + i, 0, 0);
    }
    __builtin_amdgcn_s_wait_asynccnt(0);
  }
#else
  for (int i = threadIdx.x; i < nvec; i += BLK)
    ((v4f*)sWp)[i] = ((const v4f*)Wp)[i];
#endif
  __syncthreads();

  float acc = 0.0f;
  const long base = (long)blockIdx.x * (BLK * ITEMS);
#pragma unroll 1
  for (int it = 0; it < ITEMS; ++it) {
    const long idx = base + (long)it * BLK + threadIdx.x;
    if (idx < NNZ) {
      const size_t off = (size_t)p * NNZ + (size_t)idx;
      const int i0 = s0[off];
      const int i1 = s1[off];
      const int i2 = s2[off];
      const float v = vals[off];
      const v4f* A = (const v4f*)(sWp + (size_t)i0 * RDIM);
      const v4f* B = (const v4f*)(Ul + (size_t)i1 * RDIM);
      const v4f* C = (const v4f*)(Um + (size_t)i2 * RDIM);
      float lam = 0.0f;
#pragma unroll
      for (int r = 0; r < RDIM / 4; ++r) {
        const v4f av = A[r], bv = B[r], cv = C[r];
        lam = fmaf(av.x * bv.x, cv.x, lam);
        lam = fmaf(av.y * bv.y, cv.y, lam);
        lam = fmaf(av.z * bv.z, cv.z, lam);
        lam = fmaf(av.w * bv.w, cv.w, lam);
      }
      lam = fmaxf(lam, 1e-10f);
      acc += v * __logf(lam);
    }
  }

  // wave reduce then cross-wave reduce
#pragma unroll
  for (int off = 16; off > 0; off >>= 1) acc += __shfl_down(acc, off, 32);
  const int lane = threadIdx.x & 31;
  const int wv = threadIdx.x >> 5;
  if (lane == 0) wsum[wv] = acc;
  __syncthreads();
  if (wv == 0) {
    float t = (lane < (BLK / 32)) ? wsum[lane] : 0.0f;
#pragma unroll
    for (int off = 4; off > 0; off >>= 1) t += __shfl_down(t, off, 32);
    if (lane == 0) atomicAdd(out, -t / (float)TDIM);
  }
}

// ---------------------------------------------------------------------------
extern "C" void kernel_launch(void* const* d_in, const int* in_sizes, int n_in,
                              void* d_out, int out_size, void* d_ws, size_t ws_size,
                              hipStream_t stream) {
  const int* subs0 = (const int*)d_in[0];
  const int* subs1 = (const int*)d_in[1];
  const int* subs2 = (const int*)d_in[2];
  const float* vals = (const float*)d_in[3];
  const float* Ws = (const float*)d_in[4];
  const float* Ul = (const float*)d_in[5];
  const float* Um = (const float*)d_in[6];
  float* out = (float*)d_out;
  float* sums = (float*)d_ws;  // 10 * RDIM floats

  colsum_wmma<<<dim3(2 + NPAT), dim3(128), 0, stream>>>(Ul, Um, Ws, sums);
  summ_kernel<<<dim3(1), dim3(32), 0, stream>>>(sums, out);

  const int nblk = (NNZ + BLK * ITEMS - 1) / (BLK * ITEMS);
  gather_ll<<<dim3(nblk, NPAT), dim3(BLK), 0, stream>>>(subs0, subs1, subs2,
                                                        vals, Ws, Ul, Um, out);
}